// DTDSimCell_49246095016397
// MI455X (gfx1250) — compile-verified
//
#include <hip/hip_runtime.h>

// ---------------------------------------------------------------------------
// DTD glucose-sim cell for MI455X (gfx1250, wave32).
// Dense layers (5x16, 16x16, 16x28) run on V_WMMA_F32_16X16X4_F32; tiny
// new_Z GEMM and the ODE tail stay on the VALU (hardware v_tanh_f32 for the
// two tanh calls). Layout conversion between C-layout (row-across-VGPRs) and
// A-layout (row-per-lane) bounces through a per-wave LDS tile; same-wave DS
// ordering makes this barrier-free. Persistent grid amortizes weight setup.
// Bias vectors are kept live in registers and fed directly as the WMMA C
// operand; all per-lane global addressing is 32-bit (saddr+voffset mode).
// ---------------------------------------------------------------------------

typedef float v2f __attribute__((ext_vector_type(2)));
typedef float v8f __attribute__((ext_vector_type(8)));

#define WMMA_F32X4(a, b, c) \
  __builtin_amdgcn_wmma_f32_16x16x4_f32(false, (a), false, (b), (short)0, (c), false, false)

#define WPB 8    // waves per 256-thread block
#define STR 28   // LDS activation row stride (floats); even -> 8B-aligned pairs

__device__ __forceinline__ v8f bcast8(float x) {
  v8f r;
#pragma unroll
  for (int i = 0; i < 8; ++i) r[i] = x;
  return r;
}

// Single-instruction ReLU: avoids the canonicalize that fmaxf(x,0) emits.
__device__ __forceinline__ float relu1(float x) {
  float y;
  asm("v_max_num_f32 %0, %1, 0" : "=v"(y) : "v"(x));
  return y;
}

// Hardware transcendental tanh on gfx1250 (TRANS32 pipe), libm fallback.
__device__ __forceinline__ float fast_tanh(float x) {
#if __has_builtin(__builtin_amdgcn_tanhf)
  return __builtin_amdgcn_tanhf(x);
#else
  return tanhf(x);
#endif
}

__global__ __launch_bounds__(256)
void dtd_cell_kernel(const float* __restrict__ inp, const float* __restrict__ Zg,
                     const float* __restrict__ Sg,
                     const float* __restrict__ Ag, const float* __restrict__ Bmg,
                     const float* __restrict__ W1g, const float* __restrict__ b1g,
                     const float* __restrict__ W2g, const float* __restrict__ b2g,
                     const float* __restrict__ W3g, const float* __restrict__ b3g,
                     float* __restrict__ outGp, float* __restrict__ outZ,
                     float* __restrict__ outS, long B)
{
  __shared__ float sA[25], sBm[10], sW1[80], sb1[16], sW2[256], sb2[16], sW3[448], sb3[28];
  __shared__ float sBuf[WPB][32][STR];   // per-wave activation/K staging tile

  const int tid = threadIdx.x;
  for (int i = tid; i < 25;  i += 256) sA[i]  = Ag[i];
  for (int i = tid; i < 10;  i += 256) sBm[i] = Bmg[i];
  for (int i = tid; i < 80;  i += 256) sW1[i] = W1g[i];
  for (int i = tid; i < 16;  i += 256) sb1[i] = b1g[i];
  for (int i = tid; i < 256; i += 256) sW2[i] = W2g[i];
  for (int i = tid; i < 16;  i += 256) sb2[i] = b2g[i];
  for (int i = tid; i < 448; i += 256) sW3[i] = W3g[i];
  for (int i = tid; i < 28;  i += 256) sb3[i] = b3g[i];
  __syncthreads();

  const int wave = tid >> 5;
  const int lane = tid & 31;
  const int n  = lane & 15;   // B/C column owned by this lane
  const int hi = lane >> 4;   // half-wave select (K-pair offset in A/B layout)
  float* buf = &sBuf[wave][0][0];

  // ---- Preload weights into WMMA B-layout registers (loop-invariant). ----
  // B 4x16 f32 layout (mirror of A): VGPR0 = {K0 | K2}, VGPR1 = {K1 | K3}.
  v2f Bw1[2], Bw2[4], Bw3[2][4];
#pragma unroll
  for (int j = 0; j < 2; ++j) {               // W1: K=5 padded to 8
    int r0 = 4 * j + 2 * hi;
    Bw1[j].x = (r0     < 5) ? sW1[ r0      * 16 + n] : 0.0f;
    Bw1[j].y = (r0 + 1 < 5) ? sW1[(r0 + 1) * 16 + n] : 0.0f;
  }
#pragma unroll
  for (int j = 0; j < 4; ++j) {               // W2: 16x16
    int r0 = 4 * j + 2 * hi;
    Bw2[j].x = sW2[ r0      * 16 + n];
    Bw2[j].y = sW2[(r0 + 1) * 16 + n];
  }
#pragma unroll
  for (int nt = 0; nt < 2; ++nt) {            // W3: 16x28 -> two N-tiles
#pragma unroll
    for (int j = 0; j < 4; ++j) {
      int r0 = 4 * j + 2 * hi;
      int col = nt * 16 + n;
      Bw3[nt][j].x = (col < 28) ? sW3[ r0      * 28 + col] : 0.0f;
      Bw3[nt][j].y = (col < 28) ? sW3[(r0 + 1) * 28 + col] : 0.0f;
    }
  }
  // Loop-invariant bias accumulator-init vectors (fed directly as WMMA C).
  const v8f cb1 = bcast8(sb1[n]);
  const v8f cb2 = bcast8(sb2[n]);
  v8f cb3[2];
  cb3[0] = bcast8(sb3[n]);
  cb3[1] = bcast8((16 + n < 28) ? sb3[16 + n] : 0.0f);

  const long gw     = (long)blockIdx.x * WPB + wave;
  const long stride = (long)gridDim.x * WPB * 32;

  for (long base = gw * 32; base < B; base += stride) {
    const long s = base + lane;
    const bool valid = (s < B);
    // Clamped 32-bit index: loads are always in-bounds and unconditional
    // (tail lanes read garbage that is never stored). Keeps EXEC full and
    // enables saddr + 32-bit-voffset addressing.
    const int si = (int)(valid ? s : (B - 1));

    // ---- per-lane loads ----
    float4 in4 = reinterpret_cast<const float4*>(inp)[si];
    const float* zp = Zg + si * 5;
    const float* sp = Sg + si * 9;
    float z[5], st9[9];
#pragma unroll
    for (int i = 0; i < 5; ++i) z[i] = zp[i];
#pragma unroll
    for (int i = 0; i < 9; ++i) st9[i] = sp[i];

    // ---- new_Z = Z@A + other@Bm (VALU, 35 FMA) ----
    float nz[5];
#pragma unroll
    for (int j = 0; j < 5; ++j) {
      float acc = in4.z * sBm[j] + in4.w * sBm[5 + j];
#pragma unroll
      for (int i = 0; i < 5; ++i) acc += z[i] * sA[i * 5 + j];
      nz[j] = acc;
    }
    // stage new_Z (zero-padded to K=8) as WMMA A-operand rows
#pragma unroll
    for (int k = 0; k < 5; ++k) buf[lane * STR + k] = nz[k];
#pragma unroll
    for (int k = 5; k < 8; ++k) buf[lane * STR + k] = 0.0f;

    // ---- layer 1: h1 = relu(newZ @ W1 + b1) ----
#pragma unroll
    for (int t = 0; t < 2; ++t) {
      v8f c = cb1;
#pragma unroll
      for (int j = 0; j < 2; ++j) {
        int m = t * 16 + n, kp = 4 * j + 2 * hi;
        v2f a; a.x = buf[m * STR + kp]; a.y = buf[m * STR + kp + 1];
        c = WMMA_F32X4(a, Bw1[j], c);
      }
#pragma unroll
      for (int r = 0; r < 8; ++r)
        buf[(t * 16 + 8 * hi + r) * STR + n] = relu1(c[r]);
    }

    // ---- layer 2: h2 = relu(h1 @ W2 + b2) ----
#pragma unroll
    for (int t = 0; t < 2; ++t) {
      v8f c = cb2;
#pragma unroll
      for (int j = 0; j < 4; ++j) {
        int m = t * 16 + n, kp = 4 * j + 2 * hi;
        v2f a; a.x = buf[m * STR + kp]; a.y = buf[m * STR + kp + 1];
        c = WMMA_F32X4(a, Bw2[j], c);
      }
#pragma unroll
      for (int r = 0; r < 8; ++r)
        buf[(t * 16 + 8 * hi + r) * STR + n] = relu1(c[r]);
    }

    // ---- layer 3: K = |h2 @ W3 + b3|  (N=28 -> two column tiles) ----
#pragma unroll
    for (int t = 0; t < 2; ++t) {
      v2f a[4];
#pragma unroll
      for (int j = 0; j < 4; ++j) {
        int m = t * 16 + n, kp = 4 * j + 2 * hi;
        a[j].x = buf[m * STR + kp]; a[j].y = buf[m * STR + kp + 1];
      }
#pragma unroll
      for (int nt = 0; nt < 2; ++nt) {
        v8f c = cb3[nt];
#pragma unroll
        for (int j = 0; j < 4; ++j) c = WMMA_F32X4(a[j], Bw3[nt][j], c);
        int col = nt * 16 + n;
        if (col < 28) {
#pragma unroll
          for (int r = 0; r < 8; ++r)
            buf[(t * 16 + 8 * hi + r) * STR + col] = fabsf(c[r]);
        }
      }
    }

    // ---- fetch own K row (28 f32, 16B-aligned) ----
    float K[28];
    const float4* kr = reinterpret_cast<const float4*>(&buf[lane * STR]);
#pragma unroll
    for (int q = 0; q < 7; ++q) {
      float4 v = kr[q];
      K[4 * q] = v.x; K[4 * q + 1] = v.y; K[4 * q + 2] = v.z; K[4 * q + 3] = v.w;
    }

    // ---- ODE tail (lane-scalar) ----
    const float k1 = K[0],  k2 = K[1],  m1 = K[2],  m2 = K[3],  m3 = K[4],  m4 = K[5];
    const float kgri = K[6], kmin = K[8], kmax = K[9], kabs = K[10];
    const float alpha = K[11], beta = K[12], bb = K[13], cc = K[14], D = K[15];
    const float BW = K[16], f = K[17], kp1 = K[18], kp2 = K[19], kp3 = K[20], ki = K[21];
    const float Uii = K[22], Vm0 = K[23], Vmx = K[24], Km0 = K[25], r1 = K[26], p2u = K[27];

    const float insulin = in4.x, carb = in4.y;
    const float Gp = st9[0], Gt = st9[1], Ip = st9[2], Il = st9[3];
    const float X = st9[4], XL = st9[5], Q1 = st9[6], Q2 = st9[7], Qg = st9[8];

    const float EGP   = kp1 - kp2 * Gp - kp3 * XL;
    const float Qsto  = Q1 + Q2;
    const float kempt = kmin + (kmax - kmin * 0.5f) *
        (fast_tanh(alpha * (Qsto - bb * D)) - fast_tanh(beta * (Qsto - cc * D)) + 2.0f);
    const float Ra  = f * kabs * Qg / BW;
    const float Uid = (Vm0 + Vmx * X * r1) * Gt / (Km0 + fabsf(Gt));

    const float DQ1 = -kgri * Q1 + D * carb;
    const float DQ2 = -kempt * Q2 + kgri * Q1;
    const float DQg = -kabs * Qg + kempt * Q2;
    const float DIp = -(m2 + m4) * Ip + m1 * Il + insulin;
    const float DIl = -(m1 + m3) * Il + m2 * Ip;
    const float DXL = -ki * (XL - Ip);
    const float DX  = -p2u * X + p2u * Ip;
    const float DGp = EGP + Ra - Uii - k1 * Gp + k2 * Gt;
    const float DGt = -Uid + k1 * Gp - k2 * Gt;

    if (valid) {
      outGp[si] = Gp + DGp;
      float* zo = outZ + si * 5;
      float* so = outS + si * 9;
#pragma unroll
      for (int j = 0; j < 5; ++j) zo[j] = nz[j];
      float ns[9] = {Gp + DGp, Gt + DGt, Ip + DIp, Il + DIl, X + DX,
                     XL + DXL, Q1 + DQ1, Q2 + DQ2, Qg + DQg};
#pragma unroll
      for (int j = 0; j < 9; ++j) so[j] = ns[j];
    }
  }
}

extern "C" void kernel_launch(void* const* d_in, const int* in_sizes, int n_in,
                              void* d_out, int out_size, void* d_ws, size_t ws_size,
                              hipStream_t stream) {
  const float* inp = (const float*)d_in[0];
  const float* Z   = (const float*)d_in[1];
  const float* S   = (const float*)d_in[2];
  const float* A   = (const float*)d_in[3];
  const float* Bm  = (const float*)d_in[4];
  const float* W1  = (const float*)d_in[5];
  const float* b1  = (const float*)d_in[6];
  const float* W2  = (const float*)d_in[7];
  const float* b2  = (const float*)d_in[8];
  const float* W3  = (const float*)d_in[9];
  const float* b3  = (const float*)d_in[10];

  const long B = (long)in_sizes[0] / 4;   // inputs is (B, 4)
  float* out = (float*)d_out;

  // Persistent-style grid: ~4 grid-stride iterations per wave amortizes the
  // weight-staging / B-register preload; 2048 blocks = 16K waves keeps the
  // whole chip busy.
  long need = (B + 255) / 256;
  int blocks = (int)(need < 2048 ? need : 2048);
  if (blocks < 1) blocks = 1;

  dtd_cell_kernel<<<blocks, 256, 0, stream>>>(
      inp, Z, S, A, Bm, W1, b1, W2, b2, W3, b3,
      out,            // Gp + DGp        : [B]
      out + B,        // new_Z           : [B,5]
      out + 6 * B,    // new_S           : [B,9]
      B);
}